// Encoder_17660905521340
// MI455X (gfx1250) — compile-verified
//
#include <hip/hip_runtime.h>

typedef __attribute__((ext_vector_type(2))) float v2f;
typedef __attribute__((ext_vector_type(8))) float v8f;

#define N_NODES  200000
#define N_EDGES  3200000
#define N_GRAPHS 4000
#define IN_DIM   128
#define HID      64
#define BN_EPS   1e-5f
#define GEMM_GRID 1250   // 6250 row-tiles / 1250 blocks = 5 tiles per persistent block

// -------- optional CDNA5 async global->LDS path (guarded; falls back clean) --
#if defined(__has_builtin)
#if __has_builtin(__builtin_amdgcn_global_load_async_to_lds_b128)
#define HAVE_ASYNC_LDS 1
#endif
#endif
#ifndef HAVE_ASYNC_LDS
#define HAVE_ASYNC_LDS 0
#endif

#if HAVE_ASYNC_LDS
// Builtin signature (from hipcc diagnostic): param 1 is
// '__attribute__((__vector_size__(4*sizeof(int)))) int' pointer in the global
// address space; param 2 is the LDS-side counterpart.
typedef int v4i __attribute__((vector_size(16)));
typedef __attribute__((address_space(1))) v4i g_v4i;
typedef __attribute__((address_space(3))) v4i l_v4i;
#endif

template<int N>
__device__ __forceinline__ void wait_async() {
#if HAVE_ASYNC_LDS
  asm volatile("s_wait_asynccnt %0" :: "i"(N) : "memory");
#endif
}

// Stage one 32xK row tile of A into LDS. 256 threads; 16B chunks.
template<int K>
__device__ __forceinline__ void stage_tile(float* dstLds, const float* __restrict__ A,
                                           int rowBase, int tid) {
#if HAVE_ASYNC_LDS
  const char* g = (const char*)(A + (size_t)rowBase * K);
  char* l = (char*)dstLds;
  constexpr int CH = K / 32;  // b128 chunks per thread: K=128 -> 4, K=64 -> 2
#pragma unroll
  for (int c = 0; c < CH; ++c) {
    int off = (tid + c * 256) * 16;
    __builtin_amdgcn_global_load_async_to_lds_b128(
        (g_v4i*)(g + off), (l_v4i*)(l + off), 0, 0);
  }
#else
  const float4* A4 = reinterpret_cast<const float4*>(A + (size_t)rowBase * K);
  float4* L4 = reinterpret_cast<float4*>(dstLds);
  for (int i = tid; i < (32 * K) / 4; i += 256) L4[i] = A4[i];
#endif
}

// ---------------- utility kernels ----------------

__global__ void zero_f32(float* __restrict__ p, int n) {
  int i = blockIdx.x * blockDim.x + threadIdx.x;
  if (i < n) p[i] = 0.0f;
}

__global__ void copy_f4(const float4* __restrict__ src, float4* __restrict__ dst, int n4) {
  int i = blockIdx.x * blockDim.x + threadIdx.x;
  if (i < n4) dst[i] = src[i];
}

// ---------------- edge scatter-add: agg[dst] += x[src] ----------------
template<int D>
__global__ __launch_bounds__(256) void edge_scatter(
    const int* __restrict__ srcIdx, const int* __restrict__ dstIdx,
    const float* __restrict__ x, float* __restrict__ agg) {
  const int C = D / 4;
  int tid = blockIdx.x * blockDim.x + threadIdx.x;
  if (tid >= N_EDGES * C) return;
  int e = tid / C;
  int c = tid % C;
  int s = srcIdx[e];
  int d = dstIdx[e];
  const float4 v = *reinterpret_cast<const float4*>(x + (long long)s * D + c * 4);
  float* p = agg + (long long)d * D + c * 4;
  atomicAdd(p + 0, v.x);
  atomicAdd(p + 1, v.y);
  atomicAdd(p + 2, v.z);
  atomicAdd(p + 3, v.w);
}

// ---- persistent fused GEMM + bias + ReLU (+ column sum/sumsq for BN) ------
// out[N,64] = relu(A[N,K] @ W[K,64] + bias). 256 threads = 8 waves, each wave
// owns a 16x16 tile via V_WMMA_F32_16X16X4_F32. Block loops row-tiles with
// stride gridDim; W staged once; A double-buffered (K=64) via async-to-LDS.
template<int K, bool STATS>
__global__ __launch_bounds__(256) void gin_gemm(
    const float* __restrict__ A,      // [N_NODES, K]
    const float* __restrict__ W,      // [K, 64] row-major
    const float* __restrict__ bias,   // [64]
    float* __restrict__ out,          // [N_NODES, 64]
    float* __restrict__ stats)        // [128]: colsum[64], colsumsq[64]
{
  constexpr int NBUF = (K == 64) ? 2 : 1;
  constexpr int CHUNKS = K / 32;      // async b128 chunks per thread per tile
  __shared__ float lds_w[K * 64];
  __shared__ float lds_a[NBUF][32 * K];
  __shared__ float lds_stats[128];

  const int tid = threadIdx.x;

  // Stage W once per persistent block
  for (int i = tid; i < K * 64; i += 256) lds_w[i] = W[i];
  if (STATS && tid < 128) lds_stats[tid] = 0.0f;

  const int wave  = tid >> 5;
  const int lane  = tid & 31;
  const int mTile = wave >> 2;        // 0..1
  const int nTile = wave & 3;         // 0..3
  const int ml    = lane & 15;
  const int kh    = (lane >> 4) * 2;
  const int col   = nTile * 16 + ml;
  const float bb  = bias[col];

  float gsum = 0.0f, gsq = 0.0f;
  const int NT = N_NODES / 32;        // 6250 row tiles

  int buf = 0;
  stage_tile<K>(lds_a[0], A, blockIdx.x * 32, tid);

  for (int t = blockIdx.x; t < NT; t += gridDim.x) {
    const int tn = t + (int)gridDim.x;
    if (NBUF == 2) {
      if (tn < NT) { stage_tile<K>(lds_a[buf ^ 1], A, tn * 32, tid); wait_async<CHUNKS>(); }
      else         { wait_async<0>(); }
    } else {
      wait_async<0>();
    }
    __syncthreads();  // tile t fully staged (all waves)

    v8f acc = {};
    const float* aRow = lds_a[buf] + (mTile * 16 + ml) * K;
    const float* bCol = lds_w + col;
#pragma unroll
    for (int k = 0; k < K; k += 4) {
      const int ko = k + kh;
      v2f a = *reinterpret_cast<const v2f*>(aRow + ko);   // ds_load_b64
      v2f b = { bCol[ko * 64], bCol[ko * 64 + 64] };      // ds_load x2
      acc = __builtin_amdgcn_wmma_f32_16x16x4_f32(
          false, a, false, b, (short)0, acc, false, false);
    }

    const int rBase = t * 32 + mTile * 16 + (lane >> 4) * 8;
#pragma unroll
    for (int r = 0; r < 8; ++r) {
      float v = acc[r] + bb;
      v = v > 0.0f ? v : 0.0f;
      out[(rBase + r) * 64 + col] = v;
      if (STATS) { gsum += v; gsq += v * v; }
    }
    __syncthreads();  // all reads of lds_a[buf] done before it is re-staged

    if (NBUF == 1) {
      if (tn < NT) stage_tile<K>(lds_a[0], A, tn * 32, tid);
    }
    buf ^= 1;
  }

  if (STATS) {
    atomicAdd(&lds_stats[col], gsum);        // ds float atomics
    atomicAdd(&lds_stats[64 + col], gsq);
    __syncthreads();
    if (tid < 128) atomicAdd(&stats[tid], lds_stats[tid]);
  }
}

// ---- BN finalize + fold into second-layer weights:
//  scale = g*rsqrt(var+eps); shift = bt - mean*scale
//  W2f[k][n] = scale[k]*W2[k][n];  b2f[n] = b2[n] + sum_k shift[k]*W2[k][n]
__global__ void fold_bn(const float* __restrict__ stats,
                        const float* __restrict__ g,
                        const float* __restrict__ bt,
                        const float* __restrict__ W2,
                        const float* __restrict__ b2,
                        float* __restrict__ W2f,
                        float* __restrict__ b2f) {
  __shared__ float s_scale[64];
  __shared__ float s_shift[64];
  int c = threadIdx.x;  // 64 threads
  const float invN = 1.0f / (float)N_NODES;
  float mean = stats[c] * invN;
  float var  = stats[64 + c] * invN - mean * mean;
  float sc   = g[c] * rsqrtf(var + BN_EPS);
  s_scale[c] = sc;
  s_shift[c] = bt[c] - mean * sc;
  __syncthreads();
  float acc = b2[c];
#pragma unroll 4
  for (int k = 0; k < 64; ++k) {
    float w = W2[k * 64 + c];
    W2f[k * 64 + c] = s_scale[k] * w;
    acc += s_shift[k] * w;
  }
  b2f[c] = acc;
}

// ---------------- graph readout: out[batch[n]] += h[n] ----------------
__global__ __launch_bounds__(256) void readout(
    const float* __restrict__ h, const int* __restrict__ batch,
    float* __restrict__ out) {
  int tid = blockIdx.x * blockDim.x + threadIdx.x;
  if (tid >= N_NODES * HID) return;
  int node = tid >> 6;
  int f    = tid & 63;
  atomicAdd(&out[batch[node] * HID + f], h[tid]);
}

// ---------------- host launcher ----------------
extern "C" void kernel_launch(void* const* d_in, const int* in_sizes, int n_in,
                              void* d_out, int out_size, void* d_ws, size_t ws_size,
                              hipStream_t stream) {
  (void)in_sizes; (void)n_in; (void)out_size; (void)ws_size;

  const float* x     = (const float*)d_in[0];
  const int*   ei    = (const int*)d_in[1];    // [2, N_EDGES]
  const int*   batch = (const int*)d_in[2];
  const float* w1_0  = (const float*)d_in[4];
  const float* b1_0  = (const float*)d_in[5];
  const float* g_0   = (const float*)d_in[6];
  const float* bt_0  = (const float*)d_in[7];
  const float* w2_0  = (const float*)d_in[8];
  const float* b2_0  = (const float*)d_in[9];
  const float* w1_r  = (const float*)d_in[10]; // [2,64,64]
  const float* b1_r  = (const float*)d_in[11]; // [2,64]
  const float* g_r   = (const float*)d_in[12];
  const float* bt_r  = (const float*)d_in[13];
  const float* w2_r  = (const float*)d_in[14];
  const float* b2_r  = (const float*)d_in[15];

  const int* srcIdx = ei;
  const int* dstIdx = ei + N_EDGES;

  float* ws    = (float*)d_ws;
  float* agg   = ws;                                   // N*128 (reused as N*64)
  float* hpre  = agg  + (size_t)N_NODES * IN_DIM;      // N*64
  float* h     = hpre + (size_t)N_NODES * HID;         // N*64
  float* stats = h    + (size_t)N_NODES * HID;         // 128
  float* W2f   = stats + 128;                          // 64*64
  float* b2f   = W2f + HID * HID;                      // 64

  // ---------- Layer 0 (K = 128) ----------
  {
    int n4 = N_NODES * IN_DIM / 4;
    copy_f4<<<(n4 + 255) / 256, 256, 0, stream>>>((const float4*)x, (float4*)agg, n4);
    int nes = N_EDGES * (IN_DIM / 4);
    edge_scatter<IN_DIM><<<(nes + 255) / 256, 256, 0, stream>>>(srcIdx, dstIdx, x, agg);
    zero_f32<<<1, 128, 0, stream>>>(stats, 128);
    gin_gemm<IN_DIM, true><<<GEMM_GRID, 256, 0, stream>>>(agg, w1_0, b1_0, hpre, stats);
    fold_bn<<<1, 64, 0, stream>>>(stats, g_0, bt_0, w2_0, b2_0, W2f, b2f);
    gin_gemm<HID, false><<<GEMM_GRID, 256, 0, stream>>>(hpre, W2f, b2f, h, nullptr);
  }

  // ---------- Layers 1..2 (K = 64, stacked params) ----------
  for (int l = 0; l < 2; ++l) {
    const float* W1 = w1_r + (size_t)l * HID * HID;
    const float* B1 = b1_r + (size_t)l * HID;
    const float* G  = g_r  + (size_t)l * HID;
    const float* BT = bt_r + (size_t)l * HID;
    const float* W2 = w2_r + (size_t)l * HID * HID;
    const float* B2 = b2_r + (size_t)l * HID;

    int n4 = N_NODES * HID / 4;
    copy_f4<<<(n4 + 255) / 256, 256, 0, stream>>>((const float4*)h, (float4*)agg, n4);
    int nes = N_EDGES * (HID / 4);
    edge_scatter<HID><<<(nes + 255) / 256, 256, 0, stream>>>(srcIdx, dstIdx, h, agg);
    zero_f32<<<1, 128, 0, stream>>>(stats, 128);
    gin_gemm<HID, true><<<GEMM_GRID, 256, 0, stream>>>(agg, W1, B1, hpre, stats);
    fold_bn<<<1, 64, 0, stream>>>(stats, G, BT, W2, B2, W2f, b2f);
    gin_gemm<HID, false><<<GEMM_GRID, 256, 0, stream>>>(hpre, W2f, b2f, h, nullptr);
  }

  // ---------- Readout ----------
  {
    int nOut = N_GRAPHS * HID;
    zero_f32<<<(nOut + 255) / 256, 256, 0, stream>>>((float*)d_out, nOut);
    int n = N_NODES * HID;
    readout<<<(n + 255) / 256, 256, 0, stream>>>(h, batch, (float*)d_out);
  }
}